// Sinkhorn_80126909874698
// MI455X (gfx1250) — compile-verified
//
#include <hip/hip_runtime.h>

// CDNA5 / gfx1250, wave32. One wave == one Sinkhorn system (P=64 orbitals,
// 2 per lane). C1[k] reduction done with chained V_WMMA_F32_16X16X4_F32.

typedef __attribute__((ext_vector_type(2))) float v2f;
typedef __attribute__((ext_vector_type(8))) float v8f;

#define NPART 32
#define PORB  64
#define WAVES_PER_BLOCK 8   // 256 threads/block -> 8 systems/block

__global__ __launch_bounds__(256)
void sinkhorn_wmma_kernel(const float* __restrict__ n_in,
                          const float* __restrict__ beta_p,
                          const int*   __restrict__ iters_p,
                          float* __restrict__ out, int B)
{
    const int lane = threadIdx.x & 31;
    const int sys  = blockIdx.x * WAVES_PER_BLOCK + (threadIdx.x >> 5);
    if (sys >= B) return;                      // wave-uniform; EXEC stays full

    const float beta    = beta_p[0];
    const int   n_iters = iters_p[0];
    const float invBeta = 1.0f / beta;

    // ---- load occupations: lane holds orbitals (lane) and (lane+32) ----
    const float nA = n_in[sys * PORB + lane];
    const float nB = n_in[sys * PORB + lane + 32];

    // sum(n) over 64 orbitals: local pair-add + 5-step butterfly
    float s = nA + nB;
    #pragma unroll
    for (int off = 16; off; off >>= 1) s += __shfl_xor(s, off, 32);

    const float scale = (float)NPART / s;
    const float nnA = nA * scale, nnB = nB * scale;
    const float LA = __logf(nnA / (1.0f - nnA));   // log(nn/(1-nn)) is loop-invariant
    const float LB = __logf(nnB / (1.0f - nnB));
    float epsA = -LA * invBeta;                    // eps_GC_guess
    float epsB = -LB * invBeta;

    // per-lane k exponent for the two C1 GEMM halves (columns = k index)
    const float klo   = (float)((lane & 15) + 1);  // k = 1..16
    const float khi   = klo + 16.0f;               // k = 17..32
    const int   half2 = (lane < 16) ? 0 : 2;       // which orbital slots this half feeds

    const v2f onesA = {1.0f, 1.0f};                // A = ones(16x4) => D = colsum(B)

    #pragma unroll 1
    for (int it = 0; it < n_iters; ++it) {
        // ================= C1[k] = sum_p exp(-beta*k*eps_p), k=1..32 ===========
        // GEMM view: ones(16x64) @ Bexp(64x16) accumulated in 16 K=4 chunks.
        v8f accLo = {};   // columns hold C1[1..16]
        v8f accHi = {};   // columns hold C1[17..32]
        #pragma unroll
        for (int c = 0; c < 16; ++c) {
            // chunk c covers orbitals 4c..4c+3; this lane supplies two of them
            float e0, e1;
            if (c < 8) {
                e0 = __shfl(epsA, 4 * c + half2,     32);
                e1 = __shfl(epsA, 4 * c + 1 + half2, 32);
            } else {
                e0 = __shfl(epsB, 4 * (c - 8) + half2,     32);
                e1 = __shfl(epsB, 4 * (c - 8) + 1 + half2, 32);
            }
            const float t0 = -beta * e0, t1 = -beta * e1;
            v2f blo = { __expf(t0 * klo), __expf(t1 * klo) };
            v2f bhi = { __expf(t0 * khi), __expf(t1 * khi) };
            accLo = __builtin_amdgcn_wmma_f32_16x16x4_f32(
                        false, onesA, false, blo, (short)0, accLo, false, false);
            accHi = __builtin_amdgcn_wmma_f32_16x16x4_f32(
                        false, onesA, false, bhi, (short)0, accHi, false, false);
        }
        // D VGPR0: lanes 0-15 -> C1[lane+1]; accHi lanes 16-31 -> C1[lane+1]
        const float Cv = (lane < 16) ? accLo[0] : accHi[0];

        // E[j] = C[j+1]/C[j], distributed lane j <-> E[j]; C[0] = 1
        float Cprev = __shfl_up(Cv, 1, 32);
        if (lane == 0) Cprev = 1.0f;
        const float Eml = Cv / Cprev;

        // gather E into per-lane register array (static indices after unroll)
        float Ef[NPART];
        #pragma unroll
        for (int j = 0; j < NPART; ++j) Ef[j] = __shfl(Eml, j, 32);

        const float C1_1 = Ef[0];                  // E[0] = C1[1]/1

        // ============ sequential Q recursion, reciprocal form ============
        // stores iQ[m] = 1/Q[m]; one divide per M instead of per inner step
        float iQ[NPART];
        iQ[0] = 1.0f / C1_1;
        #pragma unroll
        for (int M = 2; M <= NPART; ++M) {
            float prev = 1.0f;
            #pragma unroll
            for (int kk = 1; kk < M; ++kk)
                prev = 1.0f - Ef[M - kk] * iQ[kk - 1] * prev;   // mul + fma
            iQ[M - 1] = (float)M / (C1_1 * prev);
        }

        // ============ aux partition scan for both orbitals ============
        const float xA = __expf(-beta * epsA);
        const float xB = __expf(-beta * epsB);
        float pA = 1.0f, pB = 1.0f, pA30 = 1.0f, pB30 = 1.0f;
        #pragma unroll
        for (int k = 0; k < NPART; ++k) {
            pA = 1.0f - xA * iQ[k] * pA;
            pB = 1.0f - xB * iQ[k] * pB;
            if (k == NPART - 2) { pA30 = pA; pB30 = pB; }
        }
        const float Qp0A = pA30 * iQ[NPART - 1];   // ys[N-2] / Q[N-1]
        const float Qp0B = pB30 * iQ[NPART - 1];
        // Qp1 = pA / pB (final carries)

        // eps_new = -(L + log(Qp1) - log(Qp0)) / beta
        float enA = -(LA + __logf(pA) - __logf(Qp0A)) * invBeta;
        float enB = -(LB + __logf(pB) - __logf(Qp0B)) * invBeta;

        // normalize: eps -= sum(nn*eps_new)/sum(nn), sum(nn) == NPART
        float w = nnA * enA + nnB * enB;
        #pragma unroll
        for (int off = 16; off; off >>= 1) w += __shfl_xor(w, off, 32);
        const float corr = w * (1.0f / (float)NPART);
        epsA = enA - corr;
        epsB = enB - corr;
    }

    out[sys * PORB + lane]      = epsA;
    out[sys * PORB + lane + 32] = epsB;
}

extern "C" void kernel_launch(void* const* d_in, const int* in_sizes, int n_in,
                              void* d_out, int out_size, void* d_ws, size_t ws_size,
                              hipStream_t stream) {
    (void)n_in; (void)out_size; (void)d_ws; (void)ws_size;
    const float* n       = (const float*)d_in[0];
    const float* beta    = (const float*)d_in[1];
    const int*   n_iters = (const int*)d_in[2];
    float*       out     = (float*)d_out;

    const int B = in_sizes[0] / PORB;                       // 2048 systems
    const int blocks = (B + WAVES_PER_BLOCK - 1) / WAVES_PER_BLOCK;
    hipLaunchKernelGGL(sinkhorn_wmma_kernel, dim3(blocks), dim3(256), 0, stream,
                       n, beta, n_iters, out, B);
}